// YOLO_LOSS_36077725287010
// MI455X (gfx1250) — compile-verified
//
#include <hip/hip_runtime.h>

#define GH 14
#define GW 14
#define BBS 2
#define NC 80
#define CH (BBS*5+NC)              // 90 channels
#define CELLS (GH*GW)              // 196
#define BATCH 2048
#define MAXOBJ 32
#define PRED_PER_IMG (CELLS*CH)    // 17640 floats
#define F4_PER_IMG (PRED_PER_IMG/4)// 4410 float4 chunks

typedef __attribute__((ext_vector_type(2))) float v2f;
typedef __attribute__((ext_vector_type(8))) float v8f;

// ---------------------------------------------------------------------------
// One workgroup (256 threads = 8 wave32) per image.
// ---------------------------------------------------------------------------
__global__ __launch_bounds__(256) void yolo_loss_kernel(
    const float* __restrict__ pred,
    const float* __restrict__ gt_bbox,
    const int*   __restrict__ gt_class,
    float* __restrict__ ws)
{
  __shared__ float s_pred[PRED_PER_IMG];   // 70,560 B: whole image slab
  __shared__ float s_occ[CELLS];
  __shared__ float s_sumsq[MAXOBJ];
  __shared__ float s_obj[MAXOBJ];
  __shared__ float s_red[256];

  const int n   = blockIdx.x;
  const int tid = threadIdx.x;
  const float* gbase = pred + (size_t)n * PRED_PER_IMG;

  // ---- Phase 1: async DMA global -> LDS (CDNA5 ASYNCcnt path) -------------
  {
    unsigned lds_base = (unsigned)(uintptr_t)&s_pred[0];
    for (int i = tid; i < F4_PER_IMG; i += 256) {
      unsigned off = (unsigned)i * 16u;
      unsigned laddr = lds_base + off;
      asm volatile("global_load_async_to_lds_b128 %0, %1, %2"
                   :: "v"(laddr), "v"(off), "s"(gbase)
                   : "memory");
    }
  }

  // ---- Phase 2 (overlaps DMA): occupancy map from GT boxes ----------------
  if (tid < CELLS) s_occ[tid] = 0.0f;
  __syncthreads();
  if (tid < MAXOBJ) {
    const float* g = gt_bbox + ((size_t)n * MAXOBJ + tid) * 4;
    float cx = g[0] + g[2] * 0.5f;
    float cy = g[1] + g[3] * 0.5f;
    int col = min(max((int)floorf(cx * 0.03125f), 0), GW - 1);  // /32 exact
    int row = min(max((int)floorf(cy * 0.03125f), 0), GH - 1);
    s_occ[row * GW + col] = 1.0f;
  }

  asm volatile("s_wait_asynccnt 0x0" ::: "memory");
  __syncthreads();   // LDS slab + occupancy visible to all waves

  const int lane = tid & 31;
  const int wave = tid >> 5;

  // ---- Phase 3: per-object losses (waves 0,1 = 16 objects each) -----------
  if (wave < 2) {                      // uniform per-wave branch: EXEC == ~0
    const int mloc = lane & 15;
    const int m = wave * 16 + mloc;
    const float* g = gt_bbox + ((size_t)n * MAXOBJ + m) * 4;
    const float gx = g[0], gy = g[1], gww = g[2], ghh = g[3];
    float cx = gx + gww * 0.5f;
    float cy = gy + ghh * 0.5f;
    int col = min(max((int)floorf(cx * 0.03125f), 0), GW - 1);
    int row = min(max((int)floorf(cy * 0.03125f), 0), GH - 1);
    const float* cell = &s_pred[(row * GW + col) * CH];

    // Sum of squared class scores via V_WMMA_F32_16X16X4_F32:
    // A = 16 objects x 4 classes (f32). The register layout required for the
    // B operand of A^T equals A's own layout, so D = A * A^T with B := A, and
    // diag(D)[m] accumulates sum_k cls[m][k]^2 across the 20 K=4 chunks.
    const float* cls = cell + BBS * 5;
    const int kh = (lane < 16) ? 0 : 2;   // A/B layout: upper half holds K+2
    v8f acc = {};
#pragma unroll
    for (int c = 0; c < NC / 4; ++c) {
      v2f a;
      a.x = cls[4 * c + kh];
      a.y = cls[4 * c + kh + 1];
      acc = __builtin_amdgcn_wmma_f32_16x16x4_f32(
          false, a, false, a, (short)0, acc, false, false);
    }
    {
      // C/D layout: VGPR v holds row M=v (lanes 0-15, N=lane) and row M=v+8
      // (lanes 16-31, N=lane-16). Pull the diagonal.
      const int nn = lane & 15;
      const int mb = (lane < 16) ? 0 : 8;
#pragma unroll
      for (int v = 0; v < 8; ++v) {
        if (mb + v == nn) s_sumsq[wave * 16 + nn] = acc[v];
      }
    }

    if (lane < 16) {
      // Scalar per-object terms straight from LDS.
      float gcx = (float)col * 32.0f;
      float gcy = (float)row * 32.0f;
      float iou0 = 0.0f, iou1 = 0.0f;
      float pbx[2], pby[2], pbw[2], pbh[2], pconf[2];
#pragma unroll
      for (int b = 0; b < 2; ++b) {
        float bx = cell[b * 5 + 0], by = cell[b * 5 + 1];
        float bw = cell[b * 5 + 2], bh = cell[b * 5 + 3];
        pconf[b] = cell[b * 5 + 4];
        float pw = bw * 448.0f, ph = bh * 448.0f;
        float pcx = gcx + bx * 32.0f, pcy = gcy + by * 32.0f;
        float x1 = pcx - pw * 0.5f, y1 = pcy - ph * 0.5f;
        pbx[b] = x1; pby[b] = y1; pbw[b] = pw; pbh[b] = ph;
        float ax2 = x1 + pw, ay2 = y1 + ph;
        float bx2 = gx + gww, by2 = gy + ghh;
        float iw = fmaxf(fminf(ax2, bx2) - fmaxf(x1, gx), 0.0f);
        float ih = fmaxf(fminf(ay2, by2) - fmaxf(y1, gy), 0.0f);
        float inter = iw * ih;
        float uni = pw * ph + gww * ghh - inter;
        float iou = inter / fmaxf(uni, 1e-9f);
        if (b == 0) iou0 = iou; else iou1 = iou;
      }
      int resp = (iou1 > iou0) ? 1 : 0;   // jnp.argmax tie -> first index
      float rx = pbx[resp], ry = pby[resp];
      float rw = pbw[resp], rh = pbh[resp];
      float rc = pconf[resp];
      float dx = gx - rx, dy = gy - ry;
      float xyl = 5.0f * (dx * dx + dy * dy);
      float dsw = sqrtf(gww) - sqrtf(rw);
      float dsh = sqrtf(ghh) - sqrtf(rh);
      float whl = 5.0f * (dsw * dsw + dsh * dsh);
      int gtc = gt_class[(size_t)n * MAXOBJ + m];
      float clsg = cell[BBS * 5 + gtc];
      // class_loss = sumsq - 2*cls[gtc] + 1 ; sumsq added from s_sumsq later
      float obj = xyl + whl + 1.0f - 2.0f * clsg;
      if (m == MAXOBJ - 1) {             // conf_obj uses the LAST object only
        float d = 1.0f - rc;
        obj += d * d;
      }
      s_obj[m] = obj;
    }
  }

  // ---- Phase 4: no-object confidence penalty (one cell per thread) --------
  float local = 0.0f;
  if (tid < CELLS) {
    float occ = s_occ[tid];
    float c1 = s_pred[tid * CH + 4];
    float c2 = s_pred[tid * CH + 9];
    local = 0.5f * (1.0f - occ) * (c1 * c1 + c2 * c2);
  }

  // ---- Phase 5: deterministic in-block reduction --------------------------
  __syncthreads();   // s_obj / s_sumsq written
  s_red[tid] = local + ((tid < MAXOBJ) ? (s_obj[tid] + s_sumsq[tid]) : 0.0f);
  __syncthreads();
#pragma unroll
  for (int st = 128; st > 0; st >>= 1) {
    if (tid < st) s_red[tid] += s_red[tid + st];
    __syncthreads();
  }
  if (tid == 0) ws[n] = s_red[0];
}

// ---------------------------------------------------------------------------
// Deterministic final reduction of the 2048 per-image partials (no atomics).
// ---------------------------------------------------------------------------
__global__ __launch_bounds__(256) void yolo_reduce_kernel(
    const float* __restrict__ ws, float* __restrict__ out)
{
  __shared__ float s[256];
  int t = threadIdx.x;
  float a = 0.0f;
#pragma unroll
  for (int k = 0; k < BATCH; k += 256) a += ws[k + t];
  s[t] = a;
  __syncthreads();
#pragma unroll
  for (int st = 128; st > 0; st >>= 1) {
    if (t < st) s[t] += s[t + st];
    __syncthreads();
  }
  if (t == 0) out[0] = s[0];
}

extern "C" void kernel_launch(void* const* d_in, const int* in_sizes, int n_in,
                              void* d_out, int out_size, void* d_ws, size_t ws_size,
                              hipStream_t stream) {
  const float* pred    = (const float*)d_in[0];
  const float* gt_bbox = (const float*)d_in[1];
  const int*   gt_cls  = (const int*)d_in[2];
  float* out = (float*)d_out;
  float* ws  = (float*)d_ws;   // 2048 floats of scratch partials

  yolo_loss_kernel<<<BATCH, 256, 0, stream>>>(pred, gt_bbox, gt_cls, ws);
  yolo_reduce_kernel<<<1, 256, 0, stream>>>(ws, out);
}